// BotRGCN1_5531917877297
// MI455X (gfx1250) — compile-verified
//
#include <hip/hip_runtime.h>

// ---------------------------------------------------------------------------
// BotRGCN forward for MI455X (gfx1250, wave32).
// GEMMs use V_WMMA_F32_16X16X4_F32 (full-precision f32 matrix path).
// Edge aggregation uses fp32 global atomics (hits 192MB L2: 51MB buffers).
// ---------------------------------------------------------------------------

typedef __attribute__((ext_vector_type(2))) float v2f;
typedef __attribute__((ext_vector_type(8))) float v8f;

#define DFEAT 128
#define KC 32
#define LDS_STRIDE 36   // 32 + 4 pad: keeps 16B alignment, spreads LDS banks

// ---------------------------------------------------------------------------
// GEMM: O[N,128] = act(X[N,K] @ W[K,128] + bias)
// Block = 256 threads = 8 waves. Block tile = 128 rows x 128 cols.
// Wave w owns rows [w*16, w*16+16) x all 8 column tiles (acc = 8 x v8f).
// K staged through LDS in 32-wide chunks; W staged transposed so both A and
// B fragments for the 16x16x4 f32 WMMA are contiguous 8-byte LDS loads:
//   A frag: lane(l<16) -> A[m=l][k,k+1],  lane(l>=16) -> A[m=l-16][k+2,k+3]
//   B frag: lane(l<16) -> B[k,k+1][n=l],  lane(l>=16) -> B[k+2,k+3][n=l-16]
// ---------------------------------------------------------------------------
template<bool ACT>
__global__ __launch_bounds__(256)
void gemm128_wmma(const float* __restrict__ Xg, const float* __restrict__ Wg,
                  const float* __restrict__ bias, float* __restrict__ Og,
                  int Nrows, int K)
{
    __shared__ float As[128 * LDS_STRIDE];   // [row][k]  (chunk of X)
    __shared__ float Bs[128 * LDS_STRIDE];   // [col][k]  (W chunk, transposed)

    const int tid  = threadIdx.x;
    const int lane = tid & 31;
    const int wave = tid >> 5;        // 0..7
    const int half = lane >> 4;       // 0 / 1
    const int l16  = lane & 15;
    const int row0 = blockIdx.x * 128;

    v8f acc[8] = {};

    for (int kc = 0; kc < K; kc += KC) {
        // ---- stage A chunk: 128 rows x 32 cols, float4-vectorized ----
        #pragma unroll
        for (int i = 0; i < 4; ++i) {
            int idx4 = tid + i * 256;          // 0..1023 float4 slots
            int r    = idx4 >> 3;              // 0..127
            int q    = idx4 & 7;               // float4 within row
            int grow = row0 + r;
            float4 v = make_float4(0.f, 0.f, 0.f, 0.f);
            if (grow < Nrows) {
                const float4* p =
                    reinterpret_cast<const float4*>(Xg + (size_t)grow * K + kc);
                v = p[q];
            }
            *reinterpret_cast<float4*>(&As[r * LDS_STRIDE + q * 4]) = v;
        }
        // ---- stage W chunk transposed: W[kc+kr][c] -> Bs[c][kr] ----
        #pragma unroll
        for (int i = 0; i < 16; ++i) {
            int idx = tid + i * 256;           // 0..4095
            int kr  = idx >> 7;                // 0..31
            int c   = idx & 127;
            Bs[c * LDS_STRIDE + kr] = Wg[(size_t)(kc + kr) * DFEAT + c];
        }
        __syncthreads();

        const int arow = wave * 16 + l16;
        #pragma unroll
        for (int kt = 0; kt < 8; ++kt) {
            const int k4 = kt * 4 + half * 2;  // per-lane K slice
            v2f af;
            af.x = As[arow * LDS_STRIDE + k4 + 0];
            af.y = As[arow * LDS_STRIDE + k4 + 1];
            #pragma unroll
            for (int ct = 0; ct < 8; ++ct) {
                v2f bf;
                bf.x = Bs[(ct * 16 + l16) * LDS_STRIDE + k4 + 0];
                bf.y = Bs[(ct * 16 + l16) * LDS_STRIDE + k4 + 1];
                // D = A(16x4,f32) * B(4x16,f32) + C(16x16,f32)
                acc[ct] = __builtin_amdgcn_wmma_f32_16x16x4_f32(
                    false, af, false, bf, (short)0, acc[ct], false, false);
            }
        }
        __syncthreads();
    }

    // ---- epilogue: bias + optional LeakyReLU, D layout:
    //      acc[ct][i] -> row = row0 + wave*16 + i + half*8, col = ct*16+l16
    #pragma unroll
    for (int ct = 0; ct < 8; ++ct) {
        const int col = ct * 16 + l16;
        const float bv = bias ? bias[col] : 0.0f;
        #pragma unroll
        for (int i = 0; i < 8; ++i) {
            const int row = row0 + wave * 16 + i + half * 8;
            if (row < Nrows) {
                float v = acc[ct][i] + bv;
                if (ACT) v = (v >= 0.0f) ? v : 0.01f * v;
                Og[(size_t)row * DFEAT + col] = v;
            }
        }
    }
}

// ---------------------------------------------------------------------------
// Degree / reciprocal helpers (edges identical for both RGCN layers: do once)
// ---------------------------------------------------------------------------
__global__ void zero_kernel(float* __restrict__ p, int n)
{
    int i = blockIdx.x * blockDim.x + threadIdx.x;
    if (i < n) p[i] = 0.0f;
}

__global__ void deg_kernel(const int* __restrict__ et, const int* __restrict__ dst,
                           float* __restrict__ deg, int E, int N)
{
    int e = blockIdx.x * blockDim.x + threadIdx.x;
    if (e < E) atomicAdd(&deg[(size_t)et[e] * N + dst[e]], 1.0f);
}

__global__ void rinv_kernel(float* __restrict__ deg, int n)
{
    int i = blockIdx.x * blockDim.x + threadIdx.x;
    if (i < n) deg[i] = 1.0f / fmaxf(deg[i], 1.0f);
}

// ---------------------------------------------------------------------------
// Edge scatter: out[dst] += h[src] * (1/max(deg,1))  for edges of `rel`.
// One wave per edge; lane l covers cols [4l, 4l+4) via float4 gather +
// 4 fp32 global atomics. Mean folded in as a multiply (deg const per dst,rel).
// ---------------------------------------------------------------------------
__global__ __launch_bounds__(256)
void scatter_kernel(const float* __restrict__ h, const int* __restrict__ src,
                    const int* __restrict__ dst, const int* __restrict__ et,
                    const float* __restrict__ rinv, float* __restrict__ out,
                    int rel, int E, int N)
{
    int gid  = blockIdx.x * blockDim.x + threadIdx.x;
    int e    = gid >> 5;
    int lane = gid & 31;
    if (e >= E) return;
    if (et[e] != rel) return;
    int s = src[e], d = dst[e];
    float sc = rinv[(size_t)rel * N + d];
    const float4 v =
        *reinterpret_cast<const float4*>(h + (size_t)s * DFEAT + lane * 4);
    float* op = out + (size_t)d * DFEAT + lane * 4;
    atomicAdd(op + 0, v.x * sc);
    atomicAdd(op + 1, v.y * sc);
    atomicAdd(op + 2, v.z * sc);
    atomicAdd(op + 3, v.w * sc);
}

// ---------------------------------------------------------------------------
// Final projection: out[N,2] = X[N,128] @ W[128,2] + b.
// One wave per row; coalesced 512B row read; wave32 shuffle reduction.
// ---------------------------------------------------------------------------
__global__ __launch_bounds__(256)
void out2_kernel(const float* __restrict__ X, const float* __restrict__ W,
                 const float* __restrict__ b, float* __restrict__ out, int Nrows)
{
    const int lane = threadIdx.x & 31;
    const int wave = threadIdx.x >> 5;
    const int r = blockIdx.x * 8 + wave;
    float w0[4], w1[4];
    #pragma unroll
    for (int i = 0; i < 4; ++i) {
        int c = lane * 4 + i;
        w0[i] = W[c * 2 + 0];
        w1[i] = W[c * 2 + 1];
    }
    if (r >= Nrows) return;   // uniform per wave
    const float4 v =
        *reinterpret_cast<const float4*>(X + (size_t)r * DFEAT + lane * 4);
    float s0 = v.x * w0[0] + v.y * w0[1] + v.z * w0[2] + v.w * w0[3];
    float s1 = v.x * w1[0] + v.y * w1[1] + v.z * w1[2] + v.w * w1[3];
    #pragma unroll
    for (int off = 16; off > 0; off >>= 1) {
        s0 += __shfl_down(s0, off, 32);
        s1 += __shfl_down(s1, off, 32);
    }
    if (lane == 0) {
        out[(size_t)r * 2 + 0] = s0 + b[0];
        out[(size_t)r * 2 + 1] = s1 + b[1];
    }
}

// ---------------------------------------------------------------------------
extern "C" void kernel_launch(void* const* d_in, const int* in_sizes, int n_in,
                              void* d_out, int out_size, void* d_ws, size_t ws_size,
                              hipStream_t stream)
{
    const float* des    = (const float*)d_in[0];
    // d_in[1..3]: tweet / num_prop / cat_prop — unused by the reference source
    const int*   eidx   = (const int*)  d_in[4];   // [2,E]
    const int*   etype  = (const int*)  d_in[5];   // [E]
    const float* W_des  = (const float*)d_in[6];   // [768,128]
    const float* b_des  = (const float*)d_in[7];
    const float* W_in   = (const float*)d_in[8];   // [128,128]
    const float* b_in   = (const float*)d_in[9];
    const float* W_rel  = (const float*)d_in[10];  // [2,128,128]
    const float* W_root = (const float*)d_in[11];  // [128,128]
    const float* b_rgcn = (const float*)d_in[12];
    const float* W_out1 = (const float*)d_in[13];  // [128,128]
    const float* b_out1 = (const float*)d_in[14];
    const float* W_out2 = (const float*)d_in[15];  // [128,2]
    const float* b_out2 = (const float*)d_in[16];

    const int N = in_sizes[0] / 768;
    const int E = in_sizes[5];
    const int* srcp = eidx;
    const int* dstp = eidx + E;
    const float* W_rel0 = W_rel;
    const float* W_rel1 = W_rel + 128 * 128;

    // Workspace: 3 x [N,128] feature buffers + 2N degree reciprocals (~155MB)
    float* A   = (float*)d_ws;
    float* H   = A + (size_t)N * DFEAT;
    float* B   = H + (size_t)N * DFEAT;
    float* deg = B + (size_t)N * DFEAT;      // 2*N floats

    const int gGrid  = (N + 127) / 128;                 // GEMM blocks
    const int eGrid  = (int)(((long long)E * 32 + 255) / 256);  // scatter blocks
    const int eGrid1 = (E + 255) / 256;
    const int dGrid  = (2 * N + 255) / 256;

    // x = leaky(des @ W_des + b);  x = leaky(x @ W_in + b)
    gemm128_wmma<true ><<<gGrid, 256, 0, stream>>>(des, W_des, b_des, A, N, 768);
    gemm128_wmma<true ><<<gGrid, 256, 0, stream>>>(A,   W_in,  b_in,  B, N, 128);

    // Per-(relation,dst) mean denominators -> reciprocals (shared by both layers)
    zero_kernel<<<dGrid, 256, 0, stream>>>(deg, 2 * N);
    deg_kernel <<<eGrid1, 256, 0, stream>>>(etype, dstp, deg, E, N);
    rinv_kernel<<<dGrid, 256, 0, stream>>>(deg, 2 * N);

    // ---- RGCN layer 1: A = B@W_root + b + sum_r mean_r(B@W_r) ----
    gemm128_wmma<false><<<gGrid, 256, 0, stream>>>(B, W_root, b_rgcn, A, N, 128);
    gemm128_wmma<false><<<gGrid, 256, 0, stream>>>(B, W_rel0, nullptr, H, N, 128);
    scatter_kernel<<<eGrid, 256, 0, stream>>>(H, srcp, dstp, etype, deg, A, 0, E, N);
    gemm128_wmma<false><<<gGrid, 256, 0, stream>>>(B, W_rel1, nullptr, H, N, 128);
    scatter_kernel<<<eGrid, 256, 0, stream>>>(H, srcp, dstp, etype, deg, A, 1, E, N);

    // ---- RGCN layer 2: B = A@W_root + b + sum_r mean_r(A@W_r) ----
    gemm128_wmma<false><<<gGrid, 256, 0, stream>>>(A, W_root, b_rgcn, B, N, 128);
    gemm128_wmma<false><<<gGrid, 256, 0, stream>>>(A, W_rel0, nullptr, H, N, 128);
    scatter_kernel<<<eGrid, 256, 0, stream>>>(H, srcp, dstp, etype, deg, B, 0, E, N);
    gemm128_wmma<false><<<gGrid, 256, 0, stream>>>(A, W_rel1, nullptr, H, N, 128);
    scatter_kernel<<<eGrid, 256, 0, stream>>>(H, srcp, dstp, etype, deg, B, 1, E, N);

    // x = leaky(x @ W_out1 + b);  out = x @ W_out2 + b
    gemm128_wmma<true ><<<gGrid, 256, 0, stream>>>(B, W_out1, b_out1, A, N, 128);
    out2_kernel<<<(N + 7) / 8, 256, 0, stream>>>(A, W_out2, b_out2, (float*)d_out, N);
}